// WaveletTransform_88338887344697
// MI455X (gfx1250) — compile-verified
//
#include <hip/hip_runtime.h>

// ---------------------------------------------------------------------------
// 4-level sym4 wavelet denoise (fwd DWT -> threshold -> inverse DWT) for
// 100000 rows x 512 fp32.  One wave32 per block processes 16 rows entirely
// in LDS; every convolution stage is expressed as a Toeplitz matmul executed
// with V_WMMA_F32_16X16X4_F32 (M=16 batch rows, N=16 output samples, K=taps).
// ---------------------------------------------------------------------------

typedef float v2f __attribute__((ext_vector_type(2)));
typedef float v8f __attribute__((ext_vector_type(8)));

__constant__ float c_lo[8] = {
    -0.07576571478927333f, -0.02963552764599851f, 0.49761866763201545f,
     0.8037387518059161f,   0.29785779560527736f, -0.09921954357684722f,
    -0.012603967262037833f, 0.0322231006040427f };
__constant__ float c_hi[8] = {
    -0.0322231006040427f,  -0.012603967262037833f, 0.09921954357684722f,
     0.29785779560527736f, -0.8037387518059161f,   0.49761866763201545f,
     0.02963552764599851f, -0.07576571478927333f };

#define ROWS     16
#define PAD      4        // lead pad (zeros) so IDWT can read index -2..-1
#define EXT_W    566      // symmetric-extension scratch row stride (floats)
#define A_W      526      // ping buffer row stride
#define B_W      282      // pong buffer row stride
#define D1_W     282
#define D2_W     154
#define D3_W     90
#define D4_W     58
#define BT_W     50       // Toeplitz table row stride (transposed: [n][i])
#define EXT_FILL 560
#define THRESH   10.0f

__device__ __forceinline__ v8f wmma_f32(v2f a, v2f b, v8f c) {
    // D = A(16x4) * B(4x16) + C(16x16), fp32
    return __builtin_amdgcn_wmma_f32_16x16x4_f32(
        false, a, false, b, (short)0, c, false, false);
}

__device__ __forceinline__ void zero_range(float* p, int count, int lane) {
    for (int i = lane; i < count; i += 32) p[i] = 0.0f;
}

// zero 16 tail slots of every row (kills tile-overwrite garbage, provides
// zero padding for out-of-range K reads in the next stage)
__device__ __forceinline__ void fix_tail(float* buf, int w, int n, int lane) {
    if (lane < 16) {
        for (int r = 0; r < ROWS; ++r) buf[r * w + n + lane] = 0.0f;
    }
}

__device__ __forceinline__ void thresh_rows(float* d, int w, int n, int lane) {
    for (int r = 0; r < ROWS; ++r) {
        float* p = d + r * w;
        for (int i = lane; i < n; i += 32) {
            float v = p[i];
            p[i] = (fabsf(v) <= THRESH) ? v : 0.0f;   // keep small, zero large
        }
    }
}

// ext[i] = a[5-i] (i<6) | a[i-6] (i<n+6) | a[2n+5-i] (i<n+13) | 0
__device__ __forceinline__ void build_ext(const float* a, int aW, int n,
                                          float* ext, int lane) {
    for (int r = 0; r < ROWS; ++r) {
        const float* arow = a + r * aW;
        float* erow = ext + r * EXT_W;
        for (int i = lane; i < EXT_FILL; i += 32) {
            float v = 0.0f;
            if (i < 6)            v = arow[5 - i];
            else if (i < n + 6)   v = arow[i - 6];
            else if (i < n + 13)  v = arow[2 * n + 5 - i];
            erow[i] = v;
        }
    }
}

// Forward DWT (stride-2 conv, 8 taps) as tiled Toeplitz matmul, K = 40.
// out[m][16t+n] = sum_i ext[m][32t+i] * Bt[n][i]
__device__ __forceinline__ void dwt_tiles(const float* ext, const float* Bt,
                                          float* out, int outW, int ntiles,
                                          int lane) {
    const int m = lane & 15;
    const int half = lane >> 4;
    const float* arow = ext + m * EXT_W + 2 * half;
    const float* brow = Bt  + m * BT_W + 2 * half;     // N index == lane&15
    for (int t = 0; t < ntiles; ++t) {
        v8f acc = {};
        const float* ap = arow + 32 * t;
#pragma unroll
        for (int kk = 0; kk < 10; ++kk) {
            v2f a = *(const v2f*)(ap + 4 * kk);
            v2f b = *(const v2f*)(brow + 4 * kk);
            acc = wmma_f32(a, b, acc);
        }
        float* orow = out + (half * 8) * outW + 16 * t + m;
#pragma unroll
        for (int r = 0; r < 8; ++r) orow[r * outW] = acc[r];
    }
}

// Inverse DWT tile: rec[m][16t+n] = sum_q cA[m][m0+q]*BA[q][n]
//                                  + sum_q cD[m][m0+q]*BD[q][n],  m0 = 8t-2, K=16
__device__ __forceinline__ void idwt_tiles(const float* cA, int aW,
                                           const float* cD, int dW,
                                           const float* BtA, const float* BtD,
                                           float* out, int outW, int ntiles,
                                           int lane) {
    const int m = lane & 15;
    const int half = lane >> 4;
    const float* ar = cA + m * aW + 2 * half;
    const float* dr = cD + m * dW + 2 * half;
    const float* ba = BtA + m * BT_W + 2 * half;
    const float* bd = BtD + m * BT_W + 2 * half;
    for (int t = 0; t < ntiles; ++t) {
        const int m0 = 8 * t - 2;                     // even -> b64 aligned
        v8f acc = {};
#pragma unroll
        for (int kk = 0; kk < 4; ++kk) {
            v2f a = *(const v2f*)(ar + m0 + 4 * kk);
            v2f b = *(const v2f*)(ba + 4 * kk);
            acc = wmma_f32(a, b, acc);
        }
#pragma unroll
        for (int kk = 0; kk < 4; ++kk) {
            v2f a = *(const v2f*)(dr + m0 + 4 * kk);
            v2f b = *(const v2f*)(bd + 4 * kk);
            acc = wmma_f32(a, b, acc);
        }
        float* orow = out + (half * 8) * outW + 16 * t + m;
#pragma unroll
        for (int r = 0; r < 8; ++r) orow[r * outW] = acc[r];
    }
}

__global__ __launch_bounds__(32)
void wden_kernel(const float* __restrict__ x, float* __restrict__ out,
                 int nrows) {
    __shared__ __align__(16) float sEXT[ROWS * EXT_W];
    __shared__ __align__(16) float sA  [ROWS * A_W];
    __shared__ __align__(16) float sB  [ROWS * B_W];
    __shared__ __align__(16) float sD1 [ROWS * D1_W];
    __shared__ __align__(16) float sD2 [ROWS * D2_W];
    __shared__ __align__(16) float sD3 [ROWS * D3_W];
    __shared__ __align__(16) float sD4 [ROWS * D4_W];
    __shared__ __align__(16) float sBtLo[16 * BT_W];
    __shared__ __align__(16) float sBtHi[16 * BT_W];
    __shared__ __align__(16) float sBuLo[16 * BT_W];
    __shared__ __align__(16) float sBuHi[16 * BT_W];

    const int lane = threadIdx.x & 31;
    const int row0 = blockIdx.x * ROWS;

    // --- Toeplitz filter tables (transposed: Bt[n][i]) --------------------
    for (int idx = lane; idx < 16 * BT_W; idx += 32) {
        const int n = idx / BT_W;
        const int i = idx - n * BT_W;
        // forward:  B[i][n] = filt[7 - (i - 2n)]  for 0 <= i-2n <= 7
        const int t = i - 2 * n;
        const bool vf = (t >= 0) && (t < 8);
        sBtLo[idx] = vf ? c_lo[7 - t] : 0.0f;
        sBtHi[idx] = vf ? c_hi[7 - t] : 0.0f;
        // inverse:  BA[q][n] = filt[2q - n - 3]   for 0 <= 2q-n-3 <= 7
        const int u = 2 * i - n - 3;
        const bool vu = (i < 16) && (u >= 0) && (u < 8);
        sBuLo[idx] = vu ? c_lo[u] : 0.0f;
        sBuHi[idx] = vu ? c_hi[u] : 0.0f;
    }
    zero_range(sA,  ROWS * A_W,  lane);
    zero_range(sB,  ROWS * B_W,  lane);
    zero_range(sD1, ROWS * D1_W, lane);
    zero_range(sD2, ROWS * D2_W, lane);
    zero_range(sD3, ROWS * D3_W, lane);
    zero_range(sD4, ROWS * D4_W, lane);
    __syncthreads();

    float* A  = sA  + PAD;
    float* Bf = sB  + PAD;
    float* D1 = sD1 + PAD;
    float* D2 = sD2 + PAD;
    float* D3 = sD3 + PAD;
    float* D4 = sD4 + PAD;

    // --- coalesced load: 16 rows x 512 -> LDS (float2) --------------------
    {
        const v2f* xv = (const v2f*)x;
        for (int idx = lane; idx < ROWS * 256; idx += 32) {
            const int r = idx >> 8, c = idx & 255;
            if (row0 + r < nrows)
                *(v2f*)(A + r * A_W + 2 * c) = xv[(size_t)(row0 + r) * 256 + c];
        }
    }
    __syncthreads();

    // --- forward DWT, 4 levels -------------------------------------------
    build_ext(A, A_W, 512, sEXT, lane);  __syncthreads();
    dwt_tiles(sEXT, sBtHi, D1, D1_W, 17, lane);
    dwt_tiles(sEXT, sBtLo, Bf, B_W,  17, lane);
    __syncthreads();
    fix_tail(D1, D1_W, 259, lane);  fix_tail(Bf, B_W, 259, lane);
    __syncthreads();

    build_ext(Bf, B_W, 259, sEXT, lane); __syncthreads();
    dwt_tiles(sEXT, sBtHi, D2, D2_W, 9, lane);
    dwt_tiles(sEXT, sBtLo, A,  A_W,  9, lane);
    __syncthreads();
    fix_tail(D2, D2_W, 133, lane);  fix_tail(A, A_W, 133, lane);
    __syncthreads();

    build_ext(A, A_W, 133, sEXT, lane);  __syncthreads();
    dwt_tiles(sEXT, sBtHi, D3, D3_W, 5, lane);
    dwt_tiles(sEXT, sBtLo, Bf, B_W,  5, lane);
    __syncthreads();
    fix_tail(D3, D3_W, 70, lane);   fix_tail(Bf, B_W, 70, lane);
    __syncthreads();

    build_ext(Bf, B_W, 70, sEXT, lane);  __syncthreads();
    dwt_tiles(sEXT, sBtHi, D4, D4_W, 3, lane);
    dwt_tiles(sEXT, sBtLo, A,  A_W,  3, lane);
    __syncthreads();
    fix_tail(D4, D4_W, 38, lane);   fix_tail(A, A_W, 38, lane);

    // --- threshold detail bands (keep |d| <= 10, zero the rest) -----------
    thresh_rows(D1, D1_W, 259, lane);
    thresh_rows(D2, D2_W, 133, lane);
    thresh_rows(D3, D3_W, 70,  lane);
    thresh_rows(D4, D4_W, 38,  lane);
    __syncthreads();

    // --- inverse DWT, 4 levels (truncation to detail length built in) ----
    idwt_tiles(A,  A_W, D4, D4_W, sBuLo, sBuHi, Bf, B_W,  5, lane);   // 38 -> 70
    __syncthreads();
    fix_tail(Bf, B_W, 70, lane);
    __syncthreads();
    idwt_tiles(Bf, B_W, D3, D3_W, sBuLo, sBuHi, A,  A_W,  9, lane);   // 70 -> 133
    __syncthreads();
    fix_tail(A, A_W, 133, lane);
    __syncthreads();
    idwt_tiles(A,  A_W, D2, D2_W, sBuLo, sBuHi, Bf, B_W, 17, lane);   // 133 -> 259
    __syncthreads();
    fix_tail(Bf, B_W, 259, lane);
    __syncthreads();
    idwt_tiles(Bf, B_W, D1, D1_W, sBuLo, sBuHi, A,  A_W, 32, lane);   // 259 -> 512
    __syncthreads();

    // --- coalesced store: LDS -> 16 rows x 512 (float2) -------------------
    {
        v2f* ov = (v2f*)out;
        for (int idx = lane; idx < ROWS * 256; idx += 32) {
            const int r = idx >> 8, c = idx & 255;
            if (row0 + r < nrows)
                ov[(size_t)(row0 + r) * 256 + c] = *(const v2f*)(A + r * A_W + 2 * c);
        }
    }
}

extern "C" void kernel_launch(void* const* d_in, const int* in_sizes, int n_in,
                              void* d_out, int out_size, void* d_ws, size_t ws_size,
                              hipStream_t stream) {
    const float* x = (const float*)d_in[0];   // 100000 x 512 fp32
    // d_in[1] (edge_index) and d_in[2] (num_nodes) are unused by the reference.
    float* out = (float*)d_out;               // 100000 x 512 fp32
    const int nrows = in_sizes[0] / 512;
    const int blocks = (nrows + ROWS - 1) / ROWS;   // 6250
    hipLaunchKernelGGL(wden_kernel, dim3(blocks), dim3(32), 0, stream,
                       x, out, nrows);
}